// NonMaxSuppression_6536940225018
// MI455X (gfx1250) — compile-verified
//
#include <hip/hip_runtime.h>
#include <hip/hip_bf16.h>

typedef unsigned int u32;
typedef unsigned long long u64;

#define Hc 1024
#define Wc 1024
#define Nc (Hc * Wc)
#define TW 128
#define TH 16
#define TPITCH (TW + 2)   // 130
#define TROWS  (TH + 2)   // 18
#define NBINS  65536
#define SEL_CAP 2048
#define EQ_SORT 4096
#define KPAD_MAX 2048

#if defined(__gfx1250__) && __has_builtin(__builtin_amdgcn_tensor_load_to_lds) && __has_builtin(__builtin_amdgcn_s_wait_tensorcnt)
#define HAVE_TDM 1
#else
#define HAVE_TDM 0
#endif

#if HAVE_TDM
typedef u32 tdm_v4u __attribute__((ext_vector_type(4)));
typedef int tdm_v8i __attribute__((ext_vector_type(8)));
typedef int tdm_v4i __attribute__((ext_vector_type(4)));
#endif

// ---------------------------------------------------------------------------
// Kernel 0: zero scratch (histograms + counters). ws is poisoned once by the
// harness and never re-poisoned, so we must clear every call (deterministic).
// ---------------------------------------------------------------------------
__global__ void zero_ws_kernel(u32* __restrict__ p, unsigned long long n) {
    unsigned long long i = (unsigned long long)blockIdx.x * blockDim.x + threadIdx.x;
    unsigned long long stride = (unsigned long long)gridDim.x * blockDim.x;
    for (; i < n; i += stride) p[i] = 0u;
}

// ---------------------------------------------------------------------------
// Kernel 1: fused sigmoid + 3x3 local-argmax suppression + level-1 histogram.
// Interior blocks stage the halo tile via the Tensor Data Mover (CDNA5 TDM):
// one scalar tensor_load_to_lds per block, synchronized with s_wait_tensorcnt.
// ---------------------------------------------------------------------------
__global__ __launch_bounds__(256) void nms_kernel(const float* __restrict__ feat,
                                                  float* __restrict__ xout,
                                                  u32* __restrict__ hist1) {
    __shared__ float tile[TROWS * TPITCH];   // only LDS object -> offset 0

    const int bx  = blockIdx.x, by = blockIdx.y, b = blockIdx.z;
    const int tid = threadIdx.x;
    const int gx0 = bx * TW, gy0 = by * TH;
    const float* fb = feat + (size_t)b * Nc;

    bool used_tdm = false;
#if HAVE_TDM
    const bool interior = (gx0 > 0) && (gy0 > 0) && (gx0 + TW < Wc) && (gy0 + TH < Hc);
    if (interior) {
        used_tdm = true;
        if (tid < 32) {   // wave 0 issues the DMA (TENSORcnt is per-wave)
            const float* gsrc = fb + (size_t)(gy0 - 1) * Wc + (gx0 - 1);
            u64 ga = (u64)(uintptr_t)gsrc;
            // ---- D# group 0 (ISA 8.3): count=1, lds_addr=0, global addr, type=2
            tdm_v4u g0;
            g0.x = 1u;                                    // count=1 (valid), user mode
            g0.y = 0u;                                    // lds_addr (tile @ LDS offset 0)
            g0.z = (u32)(ga & 0xFFFFFFFFu);               // global_addr[31:0]
            g0.w = (u32)((ga >> 32) & 0x01FFFFFFu)        // global_addr[56:32]
                 | 0x80000000u;                           // type=2 ("image")
            // ---- D# group 1 (ISA 8.4): data_size=4B, dims, tile, stride
            const u32 td0 = 0x40000000u;                  // huge tensor_dim0 (tile fully in-bounds)
            const u32 td1 = 0x40000000u;                  // huge tensor_dim1
            const u32 tile0 = TPITCH;                     // 130 elements / row
            const u32 tile1 = TROWS;                      // 18 rows
            const u64 stride0 = (u64)Wc;                  // row stride in elements
            tdm_v8i g1;
            g1[0] = (int)(2u << 16);                      // workgroup_mask=0, data_size=2 (4B)
            g1[1] = (int)((td0 & 0xFFFFu) << 16);         // abar_addr=0 | tensor_dim0[15:0]
            g1[2] = (int)((td0 >> 16) | ((td1 & 0xFFFFu) << 16));
            g1[3] = (int)((td1 >> 16) | (tile0 << 16));   // tensor_dim1 hi | tile_dim0
            g1[4] = (int)tile1;                           // tile_dim1 (tile_dim2 = 0)
            g1[5] = (int)(u32)(stride0 & 0xFFFFFFFFu);    // tensor_dim0_stride[31:0]
            g1[6] = (int)(u32)(stride0 >> 32);            // stride hi16 | dim1_stride lo16 (=0)
            g1[7] = 0;
            tdm_v4i g2 = {0, 0, 0, 0};                    // 2-D tensor: groups 2/3 unused
            tdm_v4i g3 = {0, 0, 0, 0};
            tdm_v8i g4 = {0, 0, 0, 0, 0, 0, 0, 0};        // unused trailing group (6-arg form)
            __builtin_amdgcn_tensor_load_to_lds(g0, g1, g2, g3, g4, 0);
            __builtin_amdgcn_s_wait_tensorcnt(0);
        }
        __syncthreads();
    }
#endif
    if (!used_tdm) {
        // Border blocks (and non-TDM toolchains): cooperative halo load,
        // out-of-image cells get -inf (== sigmoid-space zero padding).
        for (int e = tid; e < TROWS * TPITCH; e += 256) {
            int r = e / TPITCH;
            int c = e - r * TPITCH;
            int gy = gy0 + r - 1, gx = gx0 + c - 1;
            float v = -__builtin_inff();
            if (gy >= 0 && gy < Hc && gx >= 0 && gx < Wc) v = fb[(size_t)gy * Wc + gx];
            tile[e] = v;
        }
        __syncthreads();
    }

    const int tx  = tid & (TW - 1);   // 0..127
    const int tyb = tid >> 7;         // 0..1
    float* xb = xout + (size_t)b * Nc;
    u32*   hb = hist1 + (size_t)b * NBINS;

    for (int rr = tyb; rr < TH; rr += 2) {
        const float* r0 = &tile[(rr    ) * TPITCH + tx];
        const float* r1 = &tile[(rr + 1) * TPITCH + tx];
        const float* r2 = &tile[(rr + 2) * TPITCH + tx];
        const float c  = r1[1];
        // kernel order 0..8; center=4: earlier indices must be strictly less,
        // later indices less-or-equal (matches argmax first-max tie-break).
        bool m = (r0[0] <  c) & (r0[1] <  c) & (r0[2] <  c) & (r1[0] <  c)
               & (r1[2] <= c) & (r2[0] <= c) & (r2[1] <= c) & (r2[2] <= c);
        float s = 1.0f / (1.0f + __expf(-c));
        float o = m ? s : 0.0f;
        xb[(size_t)(gy0 + rr) * Wc + gx0 + tx] = o;
        if (o > 0.0f) {
            u32 bits = __float_as_uint(o);          // o in (0,1] -> positive floats
            atomicAdd(&hb[bits >> 16], 1u);
        }
    }
}

// ---------------------------------------------------------------------------
// Kernel 2/4: radix-select level. One block per batch; finds the highest bin
// where the cumulative-from-top count (plus optional base) reaches K.
// outAbove = count strictly above the returned bin (including base).
// ---------------------------------------------------------------------------
__global__ __launch_bounds__(256) void radix_level_kernel(const u32* __restrict__ hist,
                                                          const u32* __restrict__ baseArr,
                                                          int Kp,
                                                          u32* __restrict__ outBin,
                                                          u32* __restrict__ outAbove) {
    const int b = blockIdx.x;
    const u32* h = hist + (size_t)b * NBINS;
    __shared__ u32 csum[256];
    const int t = threadIdx.x;
    const int chunk = NBINS / 256;   // 256 bins per thread
    u32 s = 0;
    for (int i = 0; i < chunk; ++i) s += h[t * chunk + i];
    csum[t] = s;
    __syncthreads();
    if (t == 0) {
        u32 cum = baseArr ? baseArr[b] : 0u;
        int bin = 0;
        u32 above = 0;
        int cj = -1;
        for (int j = 255; j >= 0; --j) {
            if (cum + csum[j] >= (u32)Kp) { cj = j; break; }
            cum += csum[j];
        }
        if (cj < 0) { bin = 0; above = cum; }           // fewer than K entries total
        else {
            above = cum;
            for (int i = chunk - 1; i >= 0; --i) {
                u32 hv = h[cj * chunk + i];
                if (cum + hv >= (u32)Kp) { bin = cj * chunk + i; above = cum; break; }
                cum += hv;
            }
        }
        outBin[b] = (u32)bin;
        outAbove[b] = above;
    }
}

// ---------------------------------------------------------------------------
// Kernel 3: level-2 histogram (low 16 bits) restricted to prefix == T1[b].
// ---------------------------------------------------------------------------
__global__ __launch_bounds__(256) void hist2_kernel(const float* __restrict__ x,
                                                    const u32* __restrict__ T1,
                                                    u32* __restrict__ hist2) {
    size_t i = (size_t)blockIdx.x * blockDim.x + threadIdx.x;
    u32 b = (u32)(i >> 20);                      // N = 1<<20
    u32 bits = __float_as_uint(x[i]);
    if ((bits >> 16) == T1[b])
        atomicAdd(&hist2[(size_t)b * NBINS + (bits & 0xFFFFu)], 1u);
}

// ---------------------------------------------------------------------------
// Kernel 5: compose exact threshold V* and remainder R = K - strictly-greater.
// ---------------------------------------------------------------------------
__global__ void compose_kernel(const u32* __restrict__ T1, const u32* __restrict__ t2,
                               const u32* __restrict__ G, int Kp, int Bv,
                               u32* __restrict__ Vstar, u32* __restrict__ Rrem) {
    int b = threadIdx.x;
    if (b < Bv) {
        Vstar[b] = (T1[b] << 16) | t2[b];
        u32 g = G[b];
        Rrem[b] = ((u32)Kp > g) ? ((u32)Kp - g) : 0u;
    }
}

// ---------------------------------------------------------------------------
// Kernel 6: collect candidates (order-independent sets; final sort fixes rank).
// ---------------------------------------------------------------------------
__global__ __launch_bounds__(256) void collect_kernel(const float* __restrict__ x,
                                                      const u32* __restrict__ Vstar,
                                                      u32* __restrict__ selCnt,
                                                      u32* __restrict__ eqCnt,
                                                      u32* __restrict__ selPairs,
                                                      u32* __restrict__ eqIdx) {
    size_t i = (size_t)blockIdx.x * blockDim.x + threadIdx.x;
    u32 b = (u32)(i >> 20);
    u32 idx = (u32)(i & (Nc - 1));
    u32 bits = __float_as_uint(x[i]);
    u32 vs = Vstar[b];
    if (bits > vs) {                 // all strictly-greater (fewer than K by construction)
        u32 p = atomicAdd(&selCnt[b], 1u);
        if (p < SEL_CAP) {
            u32* sp = selPairs + ((size_t)b * SEL_CAP + p) * 2;
            sp[0] = bits; sp[1] = idx;
        }
    } else if (bits == vs) {         // tie class: smallest R indices win
        u32 q = atomicAdd(&eqCnt[b], 1u);
        if (q < EQ_SORT) eqIdx[(size_t)b * EQ_SORT + q] = idx;
    }
}

// ---------------------------------------------------------------------------
// Kernel 7: per-batch finalize. Sort tie-class indices ascending, assemble K
// keys (~valbits<<32 | idx), bitonic sort -> exact top_k rank order, emit (i,j).
// ---------------------------------------------------------------------------
__global__ __launch_bounds__(512) void finalize_kernel(const u32* __restrict__ selPairs,
                                                       const u32* __restrict__ eqIdx,
                                                       const u32* __restrict__ selCnt,
                                                       const u32* __restrict__ eqCnt,
                                                       const u32* __restrict__ cntG,
                                                       const u32* __restrict__ Rrem,
                                                       const u32* __restrict__ Vstar,
                                                       float* __restrict__ graph,
                                                       int Kp, int KPad) {
    __shared__ u32 eqs[EQ_SORT];
    __shared__ u64 keys[KPAD_MAX];
    const int b = blockIdx.x, t = threadIdx.x, nt = blockDim.x;

    u32 m = eqCnt[b]; if (m > EQ_SORT) m = EQ_SORT;
    for (int i = t; i < EQ_SORT; i += nt)
        eqs[i] = (i < (int)m) ? eqIdx[(size_t)b * EQ_SORT + i] : 0xFFFFFFFFu;
    __syncthreads();

    // bitonic ascending sort of tie-class indices
    for (int k = 2; k <= EQ_SORT; k <<= 1) {
        for (int j = k >> 1; j > 0; j >>= 1) {
            for (int i = t; i < EQ_SORT; i += nt) {
                int ij = i ^ j;
                if (ij > i) {
                    u32 a = eqs[i], c2 = eqs[ij];
                    bool up = ((i & k) == 0);
                    if ((a > c2) == up) { eqs[i] = c2; eqs[ij] = a; }
                }
            }
            __syncthreads();
        }
    }

    u32 g = selCnt[b];
    if (g > cntG[b]) g = cntG[b];
    if (g > SEL_CAP) g = SEL_CAP;
    if (g > (u32)KPad) g = (u32)KPad;
    u32 R = Rrem[b];
    u32 vs = Vstar[b];

    for (int i = t; i < KPad; i += nt) {
        u64 key;
        if (i < (int)g) {
            const u32* sp = selPairs + ((size_t)b * SEL_CAP + i) * 2;
            key = ((u64)(~sp[0]) << 32) | sp[1];      // value desc, index asc
        } else if (i < (int)(g + R) && (i - (int)g) < EQ_SORT) {
            key = ((u64)(~vs) << 32) | eqs[i - (int)g];
        } else {
            key = ~0ull;                               // pad (sorts last)
        }
        keys[i] = key;
    }
    __syncthreads();

    for (int k = 2; k <= KPad; k <<= 1) {
        for (int j = k >> 1; j > 0; j >>= 1) {
            for (int i = t; i < KPad; i += nt) {
                int ij = i ^ j;
                if (ij > i) {
                    u64 a = keys[i], c2 = keys[ij];
                    bool up = ((i & k) == 0);
                    if ((a > c2) == up) { keys[i] = c2; keys[ij] = a; }
                }
            }
            __syncthreads();
        }
    }

    for (int i = t; i < Kp; i += nt) {
        u32 idx = (u32)(keys[i] & 0xFFFFFFFFu);
        graph[((size_t)b * Kp + i) * 2 + 0] = (float)(idx >> 10);     // row (W = 1024)
        graph[((size_t)b * Kp + i) * 2 + 1] = (float)(idx & 1023u);   // col
    }
}

// ---------------------------------------------------------------------------
extern "C" void kernel_launch(void* const* d_in, const int* in_sizes, int n_in,
                              void* d_out, int out_size, void* d_ws, size_t ws_size,
                              hipStream_t stream) {
    (void)n_in; (void)ws_size;
    const float* feat = (const float*)d_in[0];
    float* out = (float*)d_out;

    const int BN = in_sizes[0];           // B*C*H*W = 16 * 1 * 1024 * 1024
    const int Bv = BN / Nc;               // batch count (16)
    int Kp = (out_size - BN) / (Bv * 2);  // n_peaks derived from output layout
    if (Kp <= 0) Kp = 1024;
    if (Kp > KPAD_MAX) Kp = KPAD_MAX;
    int KPad = 1; while (KPad < Kp) KPad <<= 1;

    // workspace layout (u32 units)
    u32* ws = (u32*)d_ws;
    u32* hist1 = ws;
    u32* hist2 = ws + (size_t)Bv * NBINS;
    u32* meta  = ws + (size_t)2 * Bv * NBINS;
    u32* T1 = meta + 0 * Bv;
    u32* A1 = meta + 1 * Bv;
    u32* t2 = meta + 2 * Bv;
    u32* G  = meta + 3 * Bv;
    u32* Rr = meta + 4 * Bv;
    u32* Vs = meta + 5 * Bv;
    u32* sC = meta + 6 * Bv;
    u32* eC = meta + 7 * Bv;
    u32* selPairs = meta + 8 * Bv;
    u32* eqIdxBuf = selPairs + (size_t)Bv * SEL_CAP * 2;

    const unsigned long long zeroN = (unsigned long long)2 * Bv * NBINS + 8ull * Bv;
    zero_ws_kernel<<<2048, 256, 0, stream>>>(ws, zeroN);

    nms_kernel<<<dim3(Wc / TW, Hc / TH, Bv), 256, 0, stream>>>(feat, out, hist1);

    radix_level_kernel<<<Bv, 256, 0, stream>>>(hist1, nullptr, Kp, T1, A1);

    const int nBlocksPix = BN / 256;
    hist2_kernel<<<nBlocksPix, 256, 0, stream>>>(out, T1, hist2);

    radix_level_kernel<<<Bv, 256, 0, stream>>>(hist2, A1, Kp, t2, G);
    compose_kernel<<<1, 32, 0, stream>>>(T1, t2, G, Kp, Bv, Vs, Rr);

    collect_kernel<<<nBlocksPix, 256, 0, stream>>>(out, Vs, sC, eC, selPairs, eqIdxBuf);

    finalize_kernel<<<Bv, 512, 0, stream>>>(selPairs, eqIdxBuf, sC, eC, G, Rr, Vs,
                                            out + BN, Kp, KPad);
}